// Compose_41506563948878
// MI455X (gfx1250) — compile-verified
//
#include <hip/hip_runtime.h>
#include <hip/hip_bf16.h>

// Compose(left, right) = left(id + right) + right, trilinear + wrap boundary.
// Layout (B, D=3, X, Y, Z) fp32, X=Y=Z=160.
//
// Memory-bound gather op: ~300 MB streamed, ~0.5 GFLOP (~1.7 FLOP/B) -> no
// matmul structure, WMMA inapplicable. Bottlenecks: HBM stream (~13 us floor)
// and gather *address* issue. Strategy:
//   - left (98 MB) stays resident in the 192 MB L2; right/out are streamed
//     with non-temporal hints so they never evict it.
//   - z-corner pairs are contiguous when z doesn't wrap -> one b64 gather
//     instead of two b32 gathers (halves gather instructions: 24 -> 12).
//   - identity-position prefetch (global_prefetch_b8) warms gather lines.
//   - z maps to lanes: right/out accesses fully coalesced; gather addresses
//     are near-monotonic across lanes (displacements ~N(0,1)) so the TA
//     coalescer sees almost-sequential address streams.

namespace {
constexpr int DIM  = 160;
constexpr int NVOX = DIM * DIM * DIM;   // per-channel volume (4,096,000)
// 8-byte load with only 4-byte alignment guarantee (z-pair inside a row).
struct __attribute__((packed, aligned(4))) f2 { float lo, hi; };
}

__global__ __launch_bounds__(DIM) void compose_trilinear_wrap_kernel(
    const float* __restrict__ left,
    const float* __restrict__ right,
    float* __restrict__ out)
{
    const int z = threadIdx.x;      // fastest memory dim -> lanes (coalesced)
    const int y = blockIdx.x;
    const int x = blockIdx.y;
    const int b = blockIdx.z;

    const int spatial = (x * DIM + y) * DIM + z;
    const float* __restrict__ Lb = left  + (size_t)b * 3 * NVOX;
    const float* __restrict__ Rb = right + (size_t)b * 3 * NVOX + spatial;
    float*       __restrict__ Ob = out   + (size_t)b * 3 * NVOX + spatial;

    // Read-once displacement: non-temporal, keep L2 capacity for `left`.
    const float rx = __builtin_nontemporal_load(Rb);
    const float ry = __builtin_nontemporal_load(Rb + NVOX);
    const float rz = __builtin_nontemporal_load(Rb + 2 * NVOX);

    // Warm cache at the identity location; gathers land within a few voxels.
    __builtin_prefetch(Lb + spatial,            0, 3);
    __builtin_prefetch(Lb + NVOX + spatial,     0, 3);
    __builtin_prefetch(Lb + 2 * NVOX + spatial, 0, 3);

    // Absolute sample coordinates and trilinear weights.
    const float cx = (float)x + rx;
    const float cy = (float)y + ry;
    const float cz = (float)z + rz;
    const float fx = floorf(cx), fy = floorf(cy), fz = floorf(cz);
    const float wx = cx - fx,    wy = cy - fy,    wz = cz - fz;

    // Floor-mod wrap (correct for any displacement magnitude/sign).
    int x0 = (int)fx % DIM; if (x0 < 0) x0 += DIM;
    int y0 = (int)fy % DIM; if (y0 < 0) y0 += DIM;
    int z0 = (int)fz % DIM; if (z0 < 0) z0 += DIM;
    int x1 = x0 + 1; if (x1 == DIM) x1 = 0;
    int y1 = y0 + 1; if (y1 == DIM) y1 = 0;
    int z1 = z0 + 1; if (z1 == DIM) z1 = 0;

    // 4 (x,y)-corner row offsets, shared by all 3 channels.
    const int rX0 = x0 * DIM, rX1 = x1 * DIM;
    const int row[4] = { (rX0 + y0) * DIM, (rX0 + y1) * DIM,
                         (rX1 + y0) * DIM, (rX1 + y1) * DIM };

    const float ux = 1.0f - wx, uy = 1.0f - wy, uz = 1.0f - wz;
    const float wxy[4] = { ux*uy, ux*wy, wx*uy, wx*wy };

    // Gather 4 z-pairs x 3 channels.
    // Fast path (z0 <= 158): pair is contiguous and inside the z-row -> one
    // b64 gather per (channel, xy-corner): 12 gathers, never out of bounds.
    // Wrap path (z0 == 159): two b32 gathers each.
    float v0[3][4], v1[3][4];
    if (z1 == z0 + 1) {
#pragma unroll
        for (int d = 0; d < 3; ++d) {
            const float* __restrict__ L = Lb + d * NVOX + z0;
#pragma unroll
            for (int j = 0; j < 4; ++j) {
                const f2 p = *(const f2*)(L + row[j]);
                v0[d][j] = p.lo;
                v1[d][j] = p.hi;
            }
        }
    } else {
#pragma unroll
        for (int d = 0; d < 3; ++d) {
            const float* __restrict__ L = Lb + d * NVOX;
#pragma unroll
            for (int j = 0; j < 4; ++j) {
                v0[d][j] = L[row[j] + z0];
                v1[d][j] = L[row[j] + z1];
            }
        }
    }

    const float r3[3] = { rx, ry, rz };
#pragma unroll
    for (int d = 0; d < 3; ++d) {
        float acc = r3[d];                 // sampled + right
#pragma unroll
        for (int j = 0; j < 4; ++j) {
            acc = fmaf(wxy[j] * uz, v0[d][j], acc);
            acc = fmaf(wxy[j] * wz, v1[d][j], acc);
        }
        __builtin_nontemporal_store(acc, Ob + d * NVOX);   // write-once
    }
}

extern "C" void kernel_launch(void* const* d_in, const int* in_sizes, int n_in,
                              void* d_out, int out_size, void* d_ws, size_t ws_size,
                              hipStream_t stream) {
    (void)n_in; (void)out_size; (void)d_ws; (void)ws_size;
    const float* left  = (const float*)d_in[0];
    const float* right = (const float*)d_in[1];
    float* out = (float*)d_out;

    const int B = in_sizes[0] / (3 * NVOX);   // = 2 for the reference shapes
    dim3 grid(DIM /*y*/, DIM /*x*/, B /*batch*/);
    dim3 block(DIM /*z lanes, 5 wave32*/);
    compose_trilinear_wrap_kernel<<<grid, block, 0, stream>>>(left, right, out);
}